// MessageCalculationLayer_42554535969575
// MI455X (gfx1250) — compile-verified
//
#include <hip/hip_runtime.h>
#include <hip/hip_bf16.h>

typedef float v2f __attribute__((ext_vector_type(2)));
typedef float v4f __attribute__((ext_vector_type(4)));
typedef float v8f __attribute__((ext_vector_type(8)));

#define DD     128    // feature dim / output cols
#define KK     256    // contraction dim (H part + E part)
#define MTILE  128    // edges per block
#define LDSW   260    // LDS row stride in floats (256 + 4 pad -> conflict-free b64 reads)

__launch_bounds__(256, 2)
__global__ void msg_wmma_kernel(const float* __restrict__ H,
                                const float* __restrict__ E,
                                const int*   __restrict__ heads,
                                const float* __restrict__ W,
                                const float* __restrict__ bias,
                                float* __restrict__ out,
                                int M)
{
    // Full copy of W (128 x 256 fp32, row-major) with padded rows: 133,120 B.
    // 2 blocks/WGP still fit in 320 KB LDS.
    __shared__ float wlds[DD * LDSW];

    const int tid  = threadIdx.x;
    const int wave = tid >> 5;
    const int lane = tid & 31;

    // ---- Stage W into LDS via the CDNA5 async direct-to-LDS path ----
    // Global source is contiguous (W + i*16B); LDS dest is the padded
    // row-major layout. Each lane carries its own LDS byte address
    // (low 32 bits of the flat address == LDS offset per aperture rules).
    {
        const unsigned lds_base = (unsigned)(unsigned long long)(&wlds[0]);
        for (int i = tid; i < (DD * KK) / 4; i += 256) {
            const int row = i >> 6;      // 64 float4 per W row
            const int c4  = i & 63;
            const unsigned lds_off = lds_base + (unsigned)((row * LDSW + c4 * 4) * 4);
            const unsigned g_off   = (unsigned)i * 16u;
            asm volatile("global_load_async_to_lds_b128 %0, %1, %2"
                         :: "v"(lds_off), "v"(g_off), "s"(W)
                         : "memory");
        }
        // ASYNCcnt must reach 0 before we signal the barrier; the barrier
        // itself does not order async-to-LDS traffic.
        asm volatile("s_wait_asynccnt 0x0" ::: "memory");
    }
    __syncthreads();

    // ---- Per-lane fragment addressing (V_WMMA_F32_16X16X4_F32 layouts) ----
    // A (16x4): lane l holds M = l%16, K = 2*(l/16) + {0,1}  (v2f -> b64 load)
    // B (4x16): lane l holds N = l%16, K = 2*(l/16) + {0,1}
    const int  mloc  = wave * 16 + (lane & 15);
    long m = (long)blockIdx.x * MTILE + mloc;
    if (m > (long)M - 1) m = (long)M - 1;      // tail clamp: EXEC must stay all-1s for WMMA
    const int halfk = (lane >> 4) << 1;        // 0 or 2

    const int    head = heads[m];
    const float* aH   = H + (long)head * DD + halfk;   // gathered rows
    const float* aE   = E + m * DD + halfk;
    const float* bb   = wlds + (lane & 15) * LDSW + halfk;

    v8f acc[8] = {};   // 8 N-tiles of 16x16 fp32 accumulators

    // K = 0..127 : A = H[heads[m], :]  against  B = W[:, 0:128]^T
    #pragma unroll 8
    for (int kb = 0; kb < 32; ++kb) {
        const v2f a = *(const v2f*)(aH + kb * 4);
        const float* bk = bb + kb * 4;
        #pragma unroll
        for (int nt = 0; nt < 8; ++nt) {
            const v2f b = *(const v2f*)(bk + nt * 16 * LDSW);
            acc[nt] = __builtin_amdgcn_wmma_f32_16x16x4_f32(
                false, a, false, b, (short)0, acc[nt], false, false);
        }
    }
    // K = 128..255 : A = E[m, :]  against  B = W[:, 128:256]^T
    #pragma unroll 8
    for (int kb = 32; kb < 64; ++kb) {
        const v2f a = *(const v2f*)(aE + (kb - 32) * 4);
        const float* bk = bb + kb * 4;
        #pragma unroll
        for (int nt = 0; nt < 8; ++nt) {
            const v2f b = *(const v2f*)(bk + nt * 16 * LDSW);
            acc[nt] = __builtin_amdgcn_wmma_f32_16x16x4_f32(
                false, a, false, b, (short)0, acc[nt], false, false);
        }
    }

    // ---- Epilogue: add bias, store (C/D layout: row = r + 8*(lane>=16), col = lane%16) ----
    const long rbase = (long)blockIdx.x * MTILE + wave * 16 + ((lane >> 4) << 3);
    #pragma unroll
    for (int nt = 0; nt < 8; ++nt) {
        const int   col = nt * 16 + (lane & 15);
        const float bv  = bias[col];
        #pragma unroll
        for (int r = 0; r < 8; ++r) {
            const long row = rbase + r;
            if (row < M) out[row * DD + col] = acc[nt][r] + bv;
        }
    }
}

extern "C" void kernel_launch(void* const* d_in, const int* in_sizes, int n_in,
                              void* d_out, int out_size, void* d_ws, size_t ws_size,
                              hipStream_t stream)
{
    // setup_inputs order: H, E, heads, queries, W, b
    const float* H     = (const float*)d_in[0];
    const float* E     = (const float*)d_in[1];
    const int*   heads = (const int*)  d_in[2];
    const float* W     = (const float*)d_in[4];
    const float* bias  = (const float*)d_in[5];
    float*       out   = (float*)d_out;

    const int M    = in_sizes[2];                 // 800000 edges
    const int grid = (M + MTILE - 1) / MTILE;     // 6250 blocks

    msg_wmma_kernel<<<grid, 256, 0, stream>>>(H, E, heads, W, bias, out, M);
}